// GCN_29609504539480
// MI455X (gfx1250) — compile-verified
//
#include <hip/hip_runtime.h>
#include <math.h>

typedef __attribute__((ext_vector_type(2))) float v2f;
typedef __attribute__((ext_vector_type(8))) float v8f;

#define FEAT 128   // F_in == F_out == 128 for both conv layers
#define BM   64    // rows per GEMM block (4 x 16-row WMMA subtiles per wave)

// ---------------------------------------------------------------------------
// Degree / normalization kernels
// ---------------------------------------------------------------------------
__global__ __launch_bounds__(256) void gcn_deg_init(float* __restrict__ deg, int n) {
    int i = blockIdx.x * 256 + threadIdx.x;
    if (i < n) deg[i] = 1.0f;               // self loop contributes 1
}

__global__ __launch_bounds__(256) void gcn_deg_accum(const int* __restrict__ dst,
                                                     float* __restrict__ deg, int e) {
    int i = blockIdx.x * 256 + threadIdx.x;
    if (i < e) atomicAdd(&deg[dst[i]], 1.0f);
}

__global__ __launch_bounds__(256) void gcn_deg_finish(float* __restrict__ deg, int n) {
    int i = blockIdx.x * 256 + threadIdx.x;
    if (i < n) {
        float d = deg[i];
        deg[i] = (d > 0.0f) ? rsqrtf(d) : 0.0f;   // becomes dinv
    }
}

// ---------------------------------------------------------------------------
// Fused GEMM + accumulator init:
//   H[M,128]   = (relu_a ? relu(A) : A)[M,128] @ B[128,128]
//   AGG[m][n]  = bias[n] + H[m][n] * dinv[m]^2     (bias + self-loop term)
// One 256-thread block (8 wave32) per 64-row strip. Wave w owns N-tile w and
// 4 M-subtiles -> 4 back-to-back WMMAs per k-step sharing one B fragment
// (128 dynamic WMMAs/wave). fp32 WMMA: the op is bandwidth-bound, f32 is free.
// Interior blocks (781 of 782) take a fully unguarded fast path.
// ---------------------------------------------------------------------------
__global__ __launch_bounds__(256) void gcn_gemm128_wmma(
    const float* __restrict__ A, const float* __restrict__ B,
    float* __restrict__ H, float* __restrict__ AGG,
    const float* __restrict__ bias, const float* __restrict__ dinv,
    int M, int relu_a)
{
    __shared__ float As[BM][132];            // padded stride: banks = 4*row + k
    const int tid  = threadIdx.x;
    const int lane = tid & 31;
    const int wave = tid >> 5;
    const int m0   = blockIdx.x * BM;
    const bool full = (m0 + BM) <= M;        // block-uniform: no row guards needed

    // Stage the 64x128 A tile into LDS with float4 loads; fuse ReLU here.
    if (full) {
        for (int i = tid; i < BM * 32; i += 256) {
            const int r  = i >> 5;
            const int c4 = i & 31;
            float4 v = ((const float4*)(A + (size_t)(m0 + r) * FEAT))[c4];
            if (relu_a) {
                v.x = fmaxf(v.x, 0.f); v.y = fmaxf(v.y, 0.f);
                v.z = fmaxf(v.z, 0.f); v.w = fmaxf(v.w, 0.f);
            }
            ((float4*)&As[r][0])[c4] = v;
        }
    } else {
        for (int i = tid; i < BM * 32; i += 256) {
            const int r  = i >> 5;
            const int c4 = i & 31;
            float4 v = make_float4(0.f, 0.f, 0.f, 0.f);
            if (m0 + r < M)
                v = ((const float4*)(A + (size_t)(m0 + r) * FEAT))[c4];
            if (relu_a) {
                v.x = fmaxf(v.x, 0.f); v.y = fmaxf(v.y, 0.f);
                v.z = fmaxf(v.z, 0.f); v.w = fmaxf(v.w, 0.f);
            }
            ((float4*)&As[r][0])[c4] = v;
        }
    }
    __syncthreads();

    const int n0   = wave * 16;
    const int mrow = lane & 15;              // M index for A, N index for B/C
    const int half = lane >> 4;              // which K/M half this lane holds

    v8f acc0 = {0.f,0.f,0.f,0.f,0.f,0.f,0.f,0.f};
    v8f acc1 = acc0, acc2 = acc0, acc3 = acc0;

    for (int kb = 0; kb < FEAT; kb += 4) {
        const int k = kb + 2 * half;
        // B 4x16 fragment (shared by all 4 subtiles): K split across lane halves
        v2f b;
        b.x = B[(size_t)k * FEAT + n0 + mrow];
        b.y = B[(size_t)(k + 1) * FEAT + n0 + mrow];
        // A 16x4 fragments for the 4 M-subtiles (ISA layout)
        v2f a0, a1, a2, a3;
        a0.x = As[mrow     ][k]; a0.y = As[mrow     ][k + 1];
        a1.x = As[mrow + 16][k]; a1.y = As[mrow + 16][k + 1];
        a2.x = As[mrow + 32][k]; a2.y = As[mrow + 32][k + 1];
        a3.x = As[mrow + 48][k]; a3.y = As[mrow + 48][k + 1];
        acc0 = __builtin_amdgcn_wmma_f32_16x16x4_f32(false, a0, false, b, (short)0, acc0, false, false);
        acc1 = __builtin_amdgcn_wmma_f32_16x16x4_f32(false, a1, false, b, (short)0, acc1, false, false);
        acc2 = __builtin_amdgcn_wmma_f32_16x16x4_f32(false, a2, false, b, (short)0, acc2, false, false);
        acc3 = __builtin_amdgcn_wmma_f32_16x16x4_f32(false, a3, false, b, (short)0, acc3, false, false);
    }

    // Epilogue: C layout VGPR r -> M = r (lanes 0-15) / r+8 (lanes 16-31).
    // Write H and the initialized accumulator AGG = bias + H * dinv^2.
    const float bn = bias[n0 + mrow];
    v8f accs[4] = {acc0, acc1, acc2, acc3};
    if (full) {
        #pragma unroll
        for (int t = 0; t < 4; ++t) {
            const int mbase = m0 + t * 16 + 8 * half;
            #pragma unroll
            for (int r = 0; r < 8; ++r) {
                const int m = mbase + r;
                const float h = accs[t][r];
                const float w = dinv[m];
                const size_t idx = (size_t)m * FEAT + n0 + mrow;
                H[idx]   = h;
                AGG[idx] = bn + h * w * w;
            }
        }
    } else {
        #pragma unroll
        for (int t = 0; t < 4; ++t) {
            const int mbase = m0 + t * 16 + 8 * half;
            #pragma unroll
            for (int r = 0; r < 8; ++r) {
                const int m = mbase + r;
                if (m < M) {
                    const float h = accs[t][r];
                    const float w = dinv[m];
                    const size_t idx = (size_t)m * FEAT + n0 + mrow;
                    H[idx]   = h;
                    AGG[idx] = bn + h * w * w;
                }
            }
        }
    }
}

// ---------------------------------------------------------------------------
// Edge scatter: agg[dst] += h[src] * dinv[src]*dinv[dst]
// One wave per edge (32 float4 chunks): coalesced B128 gather + atomic adds
// that stay resident in the 192 MB L2 (h + agg working set ~51 MB).
// ---------------------------------------------------------------------------
__global__ __launch_bounds__(256) void gcn_edge_scatter(
    const float* __restrict__ h, const int* __restrict__ src,
    const int* __restrict__ dst, const float* __restrict__ dinv,
    float* __restrict__ agg, int e)
{
    int gid = blockIdx.x * 256 + threadIdx.x;
    int eid = gid >> 5;
    int c = gid & 31;
    if (eid >= e) return;
    int s = src[eid];
    int d = dst[eid];
    float w = dinv[s] * dinv[d];
    float4 v = ((const float4*)(h + (size_t)s * FEAT))[c];
    float* o = agg + (size_t)d * FEAT + c * 4;
    atomicAdd(o + 0, v.x * w);
    atomicAdd(o + 1, v.y * w);
    atomicAdd(o + 2, v.z * w);
    atomicAdd(o + 3, v.w * w);
}

// ---------------------------------------------------------------------------
// Readout: out[v] = sigmoid( relu(agg[v]) . Wout + bout )
// ---------------------------------------------------------------------------
__global__ __launch_bounds__(256) void gcn_readout(
    const float* __restrict__ agg, const float* __restrict__ Wout,
    const float* __restrict__ bout, float* __restrict__ out, int n)
{
    int v = blockIdx.x * 256 + threadIdx.x;
    if (v >= n) return;
    float s = bout[0];
    const float* row = agg + (size_t)v * FEAT;
    #pragma unroll 4
    for (int f = 0; f < FEAT; ++f)
        s += fmaxf(row[f], 0.f) * Wout[f];
    out[v] = 1.0f / (1.0f + expf(-s));
}

// ---------------------------------------------------------------------------
extern "C" void kernel_launch(void* const* d_in, const int* in_sizes, int n_in,
                              void* d_out, int out_size, void* d_ws, size_t ws_size,
                              hipStream_t stream)
{
    const float* x    = (const float*)d_in[0];
    const int*   ei   = (const int*)  d_in[1];   // [2, E] node indices
    const float* W1   = (const float*)d_in[2];
    const float* b1   = (const float*)d_in[3];
    const float* W2   = (const float*)d_in[4];
    const float* b2   = (const float*)d_in[5];
    const float* Wout = (const float*)d_in[6];
    const float* bout = (const float*)d_in[7];
    float* out = (float*)d_out;

    const int Nn = in_sizes[0] / FEAT;       // 50000
    const int E  = in_sizes[1] / 2;          // 800000
    const int* src = ei;
    const int* dst = ei + E;

    // Workspace layout: dinv | h | agg  (~51.4 MB total)
    char* ws = (char*)d_ws;
    float* dinv = (float*)ws;
    size_t off = ((size_t)Nn * sizeof(float) + 1023) & ~(size_t)1023;
    float* hbuf = (float*)(ws + off);
    off += (size_t)Nn * FEAT * sizeof(float);
    float* agg  = (float*)(ws + off);

    const dim3 blk(256);
    const int gN  = (Nn + 255) / 256;
    const int gE  = (E + 255) / 256;
    const int gEc = (E * 32 + 255) / 256;        // per (edge, chunk4)
    const int gM  = (Nn + BM - 1) / BM;          // GEMM row strips

    // Normalization
    gcn_deg_init  <<<gN, blk, 0, stream>>>(dinv, Nn);
    gcn_deg_accum <<<gE, blk, 0, stream>>>(dst, dinv, E);
    gcn_deg_finish<<<gN, blk, 0, stream>>>(dinv, Nn);

    // Layer 1: h = x @ W1 ; agg initialized with b1 + self-loop in epilogue
    gcn_gemm128_wmma<<<gM, blk, 0, stream>>>(x, W1, hbuf, agg, b1, dinv, Nn, 0);
    gcn_edge_scatter<<<gEc, blk, 0, stream>>>(hbuf, src, dst, dinv, agg, E);

    // Layer 2: h = relu(agg) @ W2 (ReLU fused in A-load); epilogue re-inits agg
    gcn_gemm128_wmma<<<gM, blk, 0, stream>>>(agg, W2, hbuf, agg, b2, dinv, Nn, 1);
    gcn_edge_scatter<<<gEc, blk, 0, stream>>>(hbuf, src, dst, dinv, agg, E);

    // Readout (ReLU fused)
    gcn_readout<<<gN, blk, 0, stream>>>(agg, Wout, bout, out, Nn);
}